// GCNConvBase_55164559949951
// MI455X (gfx1250) — compile-verified
//
#include <hip/hip_runtime.h>
#include <hip/hip_bf16.h>

// GCNConv (normalize=True, no self loops, edge_weight = ones)
//   deg  = segment_sum(ones, dst)
//   dis  = deg>0 ? rsqrt(max(deg,1)) : 0
//   h    = x @ W                       (fp32 WMMA 16x16x4)
//   out  = segment_sum(h[src]*dis[src]*dis[dst], dst) + b
//
// N = 50000 (= 3125 * 16, exact tiles), E = 800000, DIN = DOUT = 64.

#define GCN_N   50000
#define GCN_E   800000LL
#define GCN_D   64

typedef __attribute__((ext_vector_type(2))) float v2f;
typedef __attribute__((ext_vector_type(8))) float v8f;

// ---------------------------------------------------------------- init ------
// out[i*64+d] = b[d]; deg[i] = 0.  One thread per output element.
__global__ __launch_bounds__(256) void gcn_init_kernel(float* __restrict__ out,
                                                       float* __restrict__ deg,
                                                       const float* __restrict__ b,
                                                       long long total) {
  long long idx = (long long)blockIdx.x * blockDim.x + threadIdx.x;
  if (idx >= total) return;
  out[idx] = b[idx & (GCN_D - 1)];
  if ((idx & (GCN_D - 1)) == 0) deg[idx >> 6] = 0.0f;
}

// -------------------------------------------------------------- degree ------
// One thread per edge: deg[dst] += 1  (global_atomic_add_f32, resolves in L2).
__global__ __launch_bounds__(256) void gcn_degree_kernel(const long long* __restrict__ ei,
                                                         float* __restrict__ deg,
                                                         long long E) {
  long long e = (long long)blockIdx.x * blockDim.x + threadIdx.x;
  if (e >= E) return;
  long long d = ei[E + e];  // edge_index row 1 = dst
  atomicAdd(&deg[d], 1.0f);
}

// ----------------------------------------------------------------- dis ------
__global__ __launch_bounds__(256) void gcn_dis_kernel(const float* __restrict__ deg,
                                                      float* __restrict__ dis,
                                                      int n) {
  int i = blockIdx.x * blockDim.x + threadIdx.x;
  if (i >= n) return;
  float dg = deg[i];
  dis[i] = (dg > 0.0f) ? rsqrtf(fmaxf(dg, 1.0f)) : 0.0f;
}

// ---------------------------------------------------------------- GEMM ------
// h[N,64] = x[N,64] @ W[64,64], fp32 WMMA 16x16x4, one wave per 16x16 tile.
//
// ISA 7.12.2 32-bit layouts (wave32):
//   A 16x4 : lane m = lane&15; VGPR0 = A[m][kb], VGPR1 = A[m][kb+1], kb = (lane>>4)*2
//   B 4x16 : lane n = lane&15; VGPR0 = B[kb][n], VGPR1 = B[kb+1][n]
//   C 16x16: VGPR r = C[r + 8*(lane>>4)][lane&15]
__global__ __launch_bounds__(256) void gcn_gemm_wmma_kernel(const float* __restrict__ x,
                                                            const float* __restrict__ W,
                                                            float* __restrict__ h,
                                                            int n_tiles_m) {
  const int wave = (blockIdx.x * blockDim.x + threadIdx.x) >> 5;
  const int lane = threadIdx.x & 31;
  const int tile_m = wave >> 2;       // 4 tiles along DOUT=64
  const int tile_n = wave & 3;
  if (tile_m >= n_tiles_m) return;    // wave-uniform: EXEC stays all-ones for WMMA

  const int mn = lane & 15;           // row of A frag == col of B frag
  const int kb = (lane >> 4) << 1;    // 0 or 2

  const float* __restrict__ xrow = x + ((long long)tile_m * 16 + mn) * GCN_D;
  const float* __restrict__ wcol = W + tile_n * 16 + mn;

  v8f c = {};
#pragma unroll
  for (int k = 0; k < GCN_D; k += 4) {
    v2f a, bf;
    a.x  = xrow[k + kb];
    a.y  = xrow[k + kb + 1];
    bf.x = wcol[(k + kb) * GCN_D];
    bf.y = wcol[(k + kb + 1) * GCN_D];
    c = __builtin_amdgcn_wmma_f32_16x16x4_f32(
        /*neg_a=*/false, a, /*neg_b=*/false, bf,
        /*c_mod=*/(short)0, c, /*reuse_a=*/false, /*reuse_b=*/false);
  }

  const int col = tile_n * 16 + (lane & 15);
  const long long row0 = (long long)tile_m * 16 + ((lane >> 4) << 3);
#pragma unroll
  for (int r = 0; r < 8; ++r) {
    h[(row0 + r) * GCN_D + col] = c[r];
  }
}

// -------------------------------------------------------------- scatter -----
// 64 threads (2 waves) per edge; thread handles one feature.
// out[dst*64+f] += h[src*64+f] * dis[src]*dis[dst]
__global__ __launch_bounds__(256) void gcn_scatter_kernel(const float* __restrict__ h,
                                                          const long long* __restrict__ ei,
                                                          const float* __restrict__ dis,
                                                          float* __restrict__ out,
                                                          long long E) {
  long long idx = (long long)blockIdx.x * blockDim.x + threadIdx.x;
  long long e = idx >> 6;
  int f = (int)(idx & (GCN_D - 1));
  if (e >= E) return;
  long long s = ei[e];
  long long d = ei[E + e];
  float norm = dis[s] * dis[d];
  atomicAdd(&out[d * GCN_D + f], h[s * GCN_D + f] * norm);
}

// ---------------------------------------------------------------------------
extern "C" void kernel_launch(void* const* d_in, const int* in_sizes, int n_in,
                              void* d_out, int out_size, void* d_ws, size_t ws_size,
                              hipStream_t stream) {
  // setup_inputs order: x, edge_index(int64), edge_attr, return_attention_weights, W, b
  const float*     x  = (const float*)d_in[0];
  const long long* ei = (const long long*)d_in[1];
  const float*     W  = (const float*)d_in[4];
  const float*     b  = (const float*)d_in[5];
  float* out = (float*)d_out;

  const int       n = in_sizes[0] / GCN_D;   // 50000
  const long long E = (long long)in_sizes[2];// 800000 (edge_attr length)

  // workspace: deg[n] | dis[n] | h[n*64]   (~13.2 MB)
  float* ws  = (float*)d_ws;
  float* deg = ws;
  float* dis = ws + n;
  float* h   = ws + 2 * (size_t)n;

  const long long total_out = (long long)n * GCN_D;

  // 1) out = b (broadcast), deg = 0
  {
    int blocks = (int)((total_out + 255) / 256);
    gcn_init_kernel<<<blocks, 256, 0, stream>>>(out, deg, b, total_out);
  }
  // 2) deg[dst] += 1
  {
    int blocks = (int)((E + 255) / 256);
    gcn_degree_kernel<<<blocks, 256, 0, stream>>>(ei, deg, E);
  }
  // 3) dis = rsqrt(max(deg,1)) where deg>0
  {
    int blocks = (n + 255) / 256;
    gcn_dis_kernel<<<blocks, 256, 0, stream>>>(deg, dis, n);
  }
  // 4) h = x @ W  (fp32 WMMA; 3125*4 tiles, 8 waves/block)
  {
    int n_tiles_m = n / 16;                 // 3125 exactly
    int waves = n_tiles_m * 4;
    int blocks = (waves + 7) / 8;           // 256 threads = 8 waves
    gcn_gemm_wmma_kernel<<<blocks, 256, 0, stream>>>(x, W, h, n_tiles_m);
  }
  // 5) scatter-add messages
  {
    long long threads = E * GCN_D;
    int blocks = (int)((threads + 255) / 256);
    gcn_scatter_kernel<<<blocks, 256, 0, stream>>>(h, ei, dis, out, E);
  }
}